// Similarity_86818468921569
// MI455X (gfx1250) — compile-verified
//
#include <hip/hip_runtime.h>

#define DD 256
#define LC 1024
#define LQ 128
#define NB 32
#define ROWP 260   // padded LDS row stride (floats): bank-conflict-free fragment loads

typedef __attribute__((ext_vector_type(2))) float v2f;
typedef __attribute__((ext_vector_type(4))) float v4f;
typedef __attribute__((ext_vector_type(8))) float v8f;
typedef __attribute__((ext_vector_type(4))) int   v4i;

#define AS1 __attribute__((address_space(1)))
#define AS3 __attribute__((address_space(3)))

#if __has_builtin(__builtin_amdgcn_global_load_async_to_lds_b128)
#define HAS_ASYNC_COPY 1
#else
#define HAS_ASYNC_COPY 0
#endif

// Async copy of 16 bytes: global (AS1) -> LDS (AS3), tracked by ASYNCcnt
#define ASYNC_CP16(gp, lp)                                                     \
    __builtin_amdgcn_global_load_async_to_lds_b128((AS1 v4i*)(gp),             \
                                                   (AS3 v4i*)(lp), 0, 0)

// One wave (32 lanes) per row: out[r] = dot(X[r, 0:256], w)
__global__ void rowdot_kernel(const float* __restrict__ X, const float* __restrict__ w,
                              float* __restrict__ out, int nrows) {
    int wid  = (blockIdx.x * blockDim.x + threadIdx.x) >> 5;
    int lane = threadIdx.x & 31;
    if (wid >= nrows) return;
    const float* row = X + (size_t)wid * DD;
    float p = 0.f;
#pragma unroll
    for (int j = 0; j < DD / 32; ++j)
        p = fmaf(row[lane + 32 * j], w[lane + 32 * j], p);
#pragma unroll
    for (int off = 16; off > 0; off >>= 1)
        p += __shfl_xor(p, off, 32);
    if (lane == 0) out[wid] = p;
}

// Per workgroup: 128 context rows x 128 question cols, K = D = 256 staged in LDS.
// 8 waves in a 4x2 grid; each wave: 32x64 patch = 2x4 v8f accumulators.
__launch_bounds__(256, 1)
__global__ void simgemm_kernel(const float* __restrict__ ctx, const float* __restrict__ qst,
                               const float* __restrict__ W, const float* __restrict__ bias,
                               const float* __restrict__ cterm, const float* __restrict__ qterm,
                               float* __restrict__ out) {
    extern __shared__ float lds[];
    float* sctx = lds;                     // [128][ROWP] context tile (scaled by w3 in-LDS)
    float* sq   = lds + 128 * ROWP;        // [128][ROWP] question tile
    float* sw3  = lds + 2 * 128 * ROWP;    // [256] w3 (async path only)

    const int b    = blockIdx.z;
    const int cblk = blockIdx.x * 128;
    const int tid  = threadIdx.x;
    const float* w3 = W + 2 * DD;

    const float* ctxb = ctx + ((size_t)b * LC + cblk) * DD;
    const float* qb   = qst + (size_t)b * LQ * DD;

#if HAS_ASYNC_COPY
    // ---- async DMA: global -> LDS directly (ASYNCcnt), raw tiles ----
    for (int idx = tid; idx < 128 * (DD / 4); idx += 256) {
        int r = idx >> 6, k4 = idx & 63;
        ASYNC_CP16(ctxb + r * DD + k4 * 4, sctx + r * ROWP + k4 * 4);
        ASYNC_CP16(qb   + r * DD + k4 * 4, sq   + r * ROWP + k4 * 4);
    }
    if (tid < 64)
        ASYNC_CP16(w3 + tid * 4, sw3 + tid * 4);
#if __has_builtin(__builtin_amdgcn_s_wait_asynccnt)
    __builtin_amdgcn_s_wait_asynccnt(0);
#else
    asm volatile("s_wait_asynccnt 0" ::: "memory");
#endif
    __syncthreads();
    // ---- one-time in-LDS pass: fold w3 into the context tile ----
    for (int idx = tid; idx < 128 * (DD / 4); idx += 256) {
        int r = idx >> 6, k4 = idx & 63;
        v4f v  = *(v4f*)(sctx + r * ROWP + k4 * 4);
        v4f wv = *(const v4f*)(sw3 + k4 * 4);
        *(v4f*)(sctx + r * ROWP + k4 * 4) = v * wv;
    }
#else
    // ---- fallback staging: global -> VGPR -> LDS, fuse *w3 into context ----
    for (int idx = tid; idx < 128 * (DD / 4); idx += 256) {
        int r = idx >> 6, k4 = idx & 63;
        v4f v  = *(const v4f*)(ctxb + r * DD + k4 * 4);
        v4f wv = *(const v4f*)(w3 + k4 * 4);
        *(v4f*)(sctx + r * ROWP + k4 * 4) = v * wv;
    }
    for (int idx = tid; idx < 128 * (DD / 4); idx += 256) {
        int r = idx >> 6, k4 = idx & 63;
        *(v4f*)(sq + r * ROWP + k4 * 4) = *(const v4f*)(qb + r * DD + k4 * 4);
    }
#endif
    __syncthreads();

    // ---- fragment lane mapping (ISA 7.12.2, 32-bit operands) ----
    const int lane = tid & 31;
    const int wave = tid >> 5;
    const int mw   = (wave >> 1) * 32;   // 0,32,64,96 : wave's M offset
    const int nw   = (wave & 1) * 64;    // 0,64       : wave's N offset
    const int l15  = lane & 15;
    const int hi   = lane >> 4;          // 0/1: selects K pair (A,B) / M+8 (C,D)
    const int kh   = hi * 2;

    v8f acc[2][4] = {};
    const float* a0p = sctx + (mw +      l15) * ROWP + kh;
    const float* a1p = sctx + (mw + 16 + l15) * ROWP + kh;
    const float* bp0 = sq + (nw +  0 + l15) * ROWP + kh;
    const float* bp1 = sq + (nw + 16 + l15) * ROWP + kh;
    const float* bp2 = sq + (nw + 32 + l15) * ROWP + kh;
    const float* bp3 = sq + (nw + 48 + l15) * ROWP + kh;

#pragma unroll 4
    for (int k = 0; k < DD; k += 4) {
        v2f a0 = *(const v2f*)(a0p + k);
        v2f a1 = *(const v2f*)(a1p + k);
        v2f b0 = *(const v2f*)(bp0 + k);
        v2f b1 = *(const v2f*)(bp1 + k);
        v2f b2 = *(const v2f*)(bp2 + k);
        v2f b3 = *(const v2f*)(bp3 + k);
        acc[0][0] = __builtin_amdgcn_wmma_f32_16x16x4_f32(false, a0, false, b0, (short)0, acc[0][0], false, false);
        acc[1][0] = __builtin_amdgcn_wmma_f32_16x16x4_f32(false, a1, false, b0, (short)0, acc[1][0], false, false);
        acc[0][1] = __builtin_amdgcn_wmma_f32_16x16x4_f32(false, a0, false, b1, (short)0, acc[0][1], false, false);
        acc[1][1] = __builtin_amdgcn_wmma_f32_16x16x4_f32(false, a1, false, b1, (short)0, acc[1][1], false, false);
        acc[0][2] = __builtin_amdgcn_wmma_f32_16x16x4_f32(false, a0, false, b2, (short)0, acc[0][2], false, false);
        acc[1][2] = __builtin_amdgcn_wmma_f32_16x16x4_f32(false, a1, false, b2, (short)0, acc[1][2], false, false);
        acc[0][3] = __builtin_amdgcn_wmma_f32_16x16x4_f32(false, a0, false, b3, (short)0, acc[0][3], false, false);
        acc[1][3] = __builtin_amdgcn_wmma_f32_16x16x4_f32(false, a1, false, b3, (short)0, acc[1][3], false, false);
    }

    // ---- epilogue: + c_term + q_term + bias, store f32 ----
    const float bv = bias[0];
#pragma unroll
    for (int fm = 0; fm < 2; ++fm) {
#pragma unroll
        for (int fn = 0; fn < 4; ++fn) {
            const int cbase = cblk + mw + fm * 16 + hi * 8;
            const int qidx  = nw + fn * 16 + l15;
            const float qadd = qterm[b * LQ + qidx] + bv;
#pragma unroll
            for (int v = 0; v < 8; ++v) {
                const int c = cbase + v;
                out[((size_t)b * LC + c) * LQ + qidx] =
                    acc[fm][fn][v] + cterm[b * LC + c] + qadd;
            }
        }
    }
}

extern "C" void kernel_launch(void* const* d_in, const int* in_sizes, int n_in,
                              void* d_out, int out_size, void* d_ws, size_t ws_size,
                              hipStream_t stream) {
    const float* ctx  = (const float*)d_in[0];
    // d_in[1] = context_length (unused by reference)
    const float* qst  = (const float*)d_in[2];
    // d_in[3] = question_length (unused by reference)
    const float* W    = (const float*)d_in[4];
    const float* bias = (const float*)d_in[5];

    float* cterm = (float*)d_ws;              // NB*LC floats
    float* qterm = cterm + NB * LC;           // NB*LQ floats
    float* out   = (float*)d_out;

    // c_term = ctx . w1 ; q_term = q . w2   (one wave per row, 8 waves/block)
    rowdot_kernel<<<dim3((NB * LC) / 8), 256, 0, stream>>>(ctx, W,      cterm, NB * LC);
    rowdot_kernel<<<dim3((NB * LQ) / 8), 256, 0, stream>>>(qst, W + DD, qterm, NB * LQ);

    const size_t lds_bytes = (size_t)(2 * 128 * ROWP + 256) * sizeof(float);  // ~261 KB < 320 KB WGP LDS
    (void)hipFuncSetAttribute((const void*)simgemm_kernel,
                              hipFuncAttributeMaxDynamicSharedMemorySize, (int)lds_bytes);
    simgemm_kernel<<<dim3(LC / 128, 1, NB), 256, lds_bytes, stream>>>(
        ctx, qst, W, bias, cterm, qterm, out);
}